// OrbitalCofactorAntiequivarianceLayer_26834955665867
// MI455X (gfx1250) — compile-verified
//
#include <hip/hip_runtime.h>
#include <cstdint>

typedef __attribute__((ext_vector_type(2))) float v2f;
typedef __attribute__((ext_vector_type(8))) float v8f;
typedef __attribute__((ext_vector_type(4))) unsigned int u32x4;
typedef __attribute__((ext_vector_type(4))) int i32x4;
typedef __attribute__((ext_vector_type(8))) int i32x8;

#define BATCH 2048
#define NSPIN 32
#define DIM   256

// LDS layout (float indices)
#define EQ_OFF    0      // 32*256 = 8192 floats (eq tile, staged by TDM)
#define M_OFF     8192   // 32x33 padded orbital matrix = 1056 floats
#define ROWID_OFF 9248   // 32 (row permutation, stored as floats)
#define Z_OFF     9280   // 32 (U^T z = e0)
#define WV_OFF    9312   // 32 (L^T w = z)
#define Y_OFF     9344   // 32 (permuted -> row 0 of M^-1)
#define CS_OFF    9376   // 32 (cofactor slog sign per row)
#define CL_OFF    9408   // 32 (cofactor slog log per row)
#define SMEM_FLOATS 9440 // 37,760 bytes

__device__ __forceinline__ float sgnf(float x) {
  return (float)((x > 0.0f) - (x < 0.0f));
}

__global__ __launch_bounds__(128)
void orbital_cofactor_kernel(const float* __restrict__ eq,
                             const float* __restrict__ W1,
                             const float* __restrict__ W2,
                             float* __restrict__ out) {
  __shared__ float smem[SMEM_FLOATS];
  const int tid  = threadIdx.x;
  const int blk  = blockIdx.x;
  const int b    = blk >> 1;      // batch
  const int s    = blk & 1;       // spin half
  const int wave = tid >> 5;
  const int lane = tid & 31;
  const int half = lane >> 4;     // 0: lanes 0-15, 1: lanes 16-31
  const int lm   = lane & 15;

  const float* eq_tile = eq + ((size_t)b * 64 + (size_t)s * NSPIN) * DIM;

  // ---------- Phase 0: TDM async tensor load of eq tile (32x256 f32) into LDS
  if (wave == 0) {
    const uint64_t ga = (uint64_t)(uintptr_t)eq_tile;
    u32x4 g0;
    g0.x = 1u;                                   // count = 1 valid descriptor
    g0.y = 0u;                                   // lds_addr = 0 (eq staged at LDS base)
    g0.z = (uint32_t)ga;                         // global_addr[31:0]
    g0.w = (uint32_t)(ga >> 32) | (2u << 30);    // global_addr[56:32] | type=2
    i32x8 g1;
    g1[0] = (int)(2u << 16);                     // data_size = 2 -> 4 bytes
    g1[1] = (int)(256u << 16);                   // tensor_dim0 = 256 (bits 79:48 lo)
    g1[2] = (int)(32u << 16);                    // tensor_dim1 = 32  (bits 111:80)
    g1[3] = (int)(256u << 16);                   // tile_dim0 = 256   (bits 127:112)
    g1[4] = 32;                                  // tile_dim1 = 32
    g1[5] = 256;                                 // tensor_dim0_stride = 256
    g1[6] = 0;
    g1[7] = 0;
    i32x4 gz = {0, 0, 0, 0};
#if defined(__clang_major__) && (__clang_major__ >= 23)
    i32x8 gz8 = {0, 0, 0, 0, 0, 0, 0, 0};
    __builtin_amdgcn_tensor_load_to_lds(g0, g1, gz, gz, gz8, 0);
#else
    __builtin_amdgcn_tensor_load_to_lds(g0, g1, gz, gz, 0);
#endif
    __builtin_amdgcn_s_wait_tensorcnt(0);
  }
  __syncthreads();

  // ---------- Phase 1: M = eq_tile(32x256) @ W(256x32) via v_wmma_f32_16x16x4_f32
  {
    const float* Wp = s ? W2 : W1;
    const int m0 = (wave >> 1) << 4;   // 4 waves -> 2x2 grid of 16x16 tiles
    const int n0 = (wave & 1) << 4;
    v8f acc = {};
    const int arow = EQ_OFF + (m0 + lm) * DIM;
    for (int k0 = 0; k0 < DIM; k0 += 4) {
      // A 16x4 f32: lanes 0-15 rows m0..m0+15 K={k0,k0+1}; lanes 16-31 K={k0+2,k0+3}
      v2f a;
      a.x = smem[arow + k0 + 2 * half];
      a.y = smem[arow + k0 + 2 * half + 1];
      // B 4x16 f32: vgpr0 rows K=k0(+1), vgpr1 rows K=k0+2(+3), cols n0+lm
      v2f bf;
      bf.x = Wp[(k0 + half) * NSPIN + n0 + lm];
      bf.y = Wp[(k0 + 2 + half) * NSPIN + n0 + lm];
      acc = __builtin_amdgcn_wmma_f32_16x16x4_f32(false, a, false, bf,
                                                  (short)0, acc, false, false);
    }
#pragma unroll
    for (int v = 0; v < 8; ++v)  // D layout: vgpr v -> row v (lanes 0-15) / v+8 (16-31)
      smem[M_OFF + (m0 + v + 8 * half) * 33 + (n0 + lm)] = acc[v];
  }
  __syncthreads();

  // ---------- Phase 2: wave 0 -> LU w/ partial pivoting; cofactor = det(M)*invrow0
  if (wave == 0) {
    const int l = lane;
    float* Mm = smem + M_OFF;
    const float col0 = Mm[l * 33 + 0];   // save M[:,0] before LU destroys it
    smem[ROWID_OFF + l] = (float)l;
    float pivsign = 1.0f;
    for (int k = 0; k < 32; ++k) {
      // pivot argmax over column k (rows >= k)
      float v = (l >= k) ? fabsf(Mm[l * 33 + k]) : -1.0f;
      int idx = l;
      for (int off = 16; off; off >>= 1) {
        float ov = __shfl_xor(v, off, 32);
        int   oi = __shfl_xor(idx, off, 32);
        if (ov > v) { v = ov; idx = oi; }
      }
      const int p = idx;                 // uniform across wave
      if (p != k) {
        float t1 = Mm[k * 33 + l], t2 = Mm[p * 33 + l];
        Mm[k * 33 + l] = t2;
        Mm[p * 33 + l] = t1;
        if (l == 0) {
          float r1 = smem[ROWID_OFF + k], r2 = smem[ROWID_OFF + p];
          smem[ROWID_OFF + k] = r2;
          smem[ROWID_OFF + p] = r1;
        }
        pivsign = -pivsign;
      }
      const float dk = Mm[k * 33 + k];
      if (l > k) {                       // lane-per-row rank-1 update
        const float m = Mm[l * 33 + k] / dk;
        Mm[l * 33 + k] = m;
        for (int j = k + 1; j < 32; ++j)
          Mm[l * 33 + j] -= m * Mm[k * 33 + j];
      }
    }
    // slogdet from U diagonal
    const float dl = Mm[l * 33 + l];
    float dlog = __logf(fabsf(dl));
    float dsgn = sgnf(dl);
    for (int off = 16; off; off >>= 1) {
      dlog += __shfl_xor(dlog, off, 32);
      dsgn *= __shfl_xor(dsgn, off, 32);
    }
    dsgn *= pivsign;
    // Solve M^T y = e0 with P M = L U => U^T z = e0 ; L^T w = z ; y[rowid[k]] = w[k]
    if (l == 0) {
      float* z = smem + Z_OFF;
      for (int j = 0; j < 32; ++j) {
        float a0 = (j == 0) ? 1.0f : 0.0f;
        for (int i = 0; i < j; ++i) a0 -= Mm[i * 33 + j] * z[i];
        z[j] = a0 / Mm[j * 33 + j];
      }
      float* wv = smem + WV_OFF;
      for (int j = 31; j >= 0; --j) {
        float a0 = z[j];
        for (int i = j + 1; i < 32; ++i) a0 -= Mm[i * 33 + j] * wv[i];
        wv[j] = a0;
      }
    }
    const int dest = (int)smem[ROWID_OFF + l];
    smem[Y_OFF + dest] = smem[WV_OFF + l];
    const float yi = smem[Y_OFF + l];
    // row i: M[i,0] * (-1)^i det(minor_i) == M[i,0] * det(M) * M^-1[0,i]
    smem[CS_OFF + l] = sgnf(col0) * dsgn * sgnf(yi);
    smem[CL_OFF + l] = __logf(fabsf(col0)) + dlog + __logf(fabsf(yi));
  }
  __syncthreads();

  // ---------- Phase 3: streaming slog epilogue (memory-bound part)
  {
    const size_t chunk = (size_t)BATCH * NSPIN * DIM;  // 16,777,216 per output
    float* outS = out + (size_t)(2 * s) * chunk + (size_t)b * NSPIN * DIM;
    float* outL = out + (size_t)(2 * s + 1) * chunk + (size_t)b * NSPIN * DIM;
    const float4* eqv = (const float4*)(smem + EQ_OFF);
#pragma unroll
    for (int j = 0; j < 16; ++j) {
      const int c = tid + 128 * j;   // float4 chunk id, 0..2047
      const int e = 4 * c;
      const int r = e >> 8;          // row (never crossed by a float4)
      const float4 x = eqv[c];
      const float cs = smem[CS_OFF + r];
      const float cl = smem[CL_OFF + r];
      float4 so, lo;
      so.x = sgnf(x.x) * cs;  lo.x = __logf(fabsf(x.x)) + cl;
      so.y = sgnf(x.y) * cs;  lo.y = __logf(fabsf(x.y)) + cl;
      so.z = sgnf(x.z) * cs;  lo.z = __logf(fabsf(x.z)) + cl;
      so.w = sgnf(x.w) * cs;  lo.w = __logf(fabsf(x.w)) + cl;
      *(float4*)(outS + e) = so;
      *(float4*)(outL + e) = lo;
    }
  }
}

extern "C" void kernel_launch(void* const* d_in, const int* in_sizes, int n_in,
                              void* d_out, int out_size, void* d_ws, size_t ws_size,
                              hipStream_t stream) {
  (void)in_sizes; (void)n_in; (void)d_ws; (void)ws_size; (void)out_size;
  const float* eq = (const float*)d_in[0];
  const float* W1 = (const float*)d_in[1];
  const float* W2 = (const float*)d_in[2];
  float* out = (float*)d_out;
  orbital_cofactor_kernel<<<dim3(BATCH * 2), dim3(128), 0, stream>>>(eq, W1, W2, out);
}